// SS2D_11759620456500
// MI455X (gfx1250) — compile-verified
//
#include <hip/hip_runtime.h>
#include <math.h>

typedef float v2f __attribute__((ext_vector_type(2)));
typedef float v8f __attribute__((ext_vector_type(8)));

#define D_MODEL 96
#define D_STATE 16
#define D_INNER 192
#define DT_RANK 6
#define BSZ 2
#define HSZ 56
#define WSZ 56
#define LSEQ (HSZ*WSZ)          // 3136
#define M1 (BSZ*LSEQ)           // 6272 rows of (B,H,W)
#define GRP 8                   // 4 dirs * B
#define MG (GRP*LSEQ)           // 25088 rows of (G,L)
#define XDBL_LD 48              // 38 padded to 48 (3 WMMA n-tiles)

// ---------------- WMMA f32 16x16x4 helper ----------------
static __device__ __forceinline__ v8f wmma4(v2f a, v2f b, v8f c) {
  // D = A(16x4,f32) * B(4x16,f32) + C(16x16,f32)
  return __builtin_amdgcn_wmma_f32_16x16x4_f32(false, a, false, b, (short)0, c,
                                               false, false);
}

// direction gather: sequence pos (r,s) of group g -> spatial (rp,sp) of xm_conv
static __device__ __forceinline__ int dir_off(int dir, int b, int r, int s, int c) {
  int rp, sp;
  if      (dir == 0) { rp = r; sp = s; }
  else if (dir == 1) { rp = r; sp = WSZ - 1 - s; }
  else if (dir == 2) { rp = s; sp = r; }
  else               { rp = s; sp = WSZ - 1 - r; }
  return ((b * HSZ + rp) * WSZ + sp) * D_INNER + c;
}

// x_dbl column permutation: dtr at 0..5, B_n at 6+2n, C_n at 7+2n, pad cols as-is.
static __device__ __forceinline__ int xdbl_pos(int nn) {
  return (nn < DT_RANK)               ? nn
       : (nn < DT_RANK + D_STATE)     ? (DT_RANK + 2 * (nn - DT_RANK))
       : (nn < DT_RANK + 2 * D_STATE) ? (DT_RANK + 1 + 2 * (nn - DT_RANK - D_STATE))
       : nn;
}

// ---------------- 1) xz = x @ W_in  (6272x96 @ 96x384) ----------------
__global__ void k_gemm_in(const float* __restrict__ X, const float* __restrict__ Win,
                          float* __restrict__ xm, float* __restrict__ z) {
  const int lane = threadIdx.x & 31;
  const int wave = threadIdx.x >> 5;
  const int half = lane >> 4;
  const int lr   = lane & 15;
  const int n0 = blockIdx.x * 64;
  const int m0 = (blockIdx.y * 8 + wave) * 16;

  v8f acc[4] = {};
  const float* arow = X + (size_t)(m0 + lr) * D_MODEL;
  for (int k0 = 0; k0 < D_MODEL; k0 += 4) {
    v2f a = *(const v2f*)(arow + k0 + 2 * half);
    const int kk = k0 + 2 * half;
    const float* bp = Win + (size_t)kk * (2 * D_INNER) + n0 + lr;
#pragma unroll
    for (int t = 0; t < 4; ++t) {
      v2f bv = { bp[t * 16], bp[2 * D_INNER + t * 16] };
      acc[t] = wmma4(a, bv, acc[t]);
    }
  }
#pragma unroll
  for (int t = 0; t < 4; ++t) {
    const int col = n0 + t * 16 + lr;
    float* outp = (col < D_INNER) ? (xm + col) : (z + (col - D_INNER));
#pragma unroll
    for (int v = 0; v < 8; ++v) {
      const int m = m0 + v + 8 * half;
      outp[(size_t)m * D_INNER] = acc[t][v];
    }
  }
}

// ---------------- 2) depthwise 3x3 conv + bias ----------------
__global__ void k_dwconv(const float* __restrict__ xm, const float* __restrict__ cw,
                         const float* __restrict__ cb, float* __restrict__ out) {
  const int idx = blockIdx.x * 256 + threadIdx.x;   // over M1*D_INNER
  const int c = idx % D_INNER;
  const int m = idx / D_INNER;
  const int s = m % WSZ;
  const int r = (m / WSZ) % HSZ;
  const int b = m / LSEQ;
  float acc = cb[c];
#pragma unroll
  for (int dr = 0; dr < 3; ++dr)
#pragma unroll
    for (int ds = 0; ds < 3; ++ds) {
      const int rr = r + dr - 1, ss = s + ds - 1;
      if (rr >= 0 && rr < HSZ && ss >= 0 && ss < WSZ)
        acc += xm[((b * HSZ + rr) * WSZ + ss) * D_INNER + c] *
               cw[(dr * 3 + ds) * D_INNER + c];
    }
  out[idx] = acc;
}

// ---------------- 3) x_dbl = xs @ W_xproj  (25088x192 @ 192x38, N pad 48) ----
// A rows gathered from xm_conv via the 4-direction index map; stores use the
// B/C-interleaved column permutation so the scan can fetch {B,C} as one b64.
__global__ void k_gemm_xproj(const float* __restrict__ xmconv,
                             const float* __restrict__ Wx,
                             float* __restrict__ xdbl) {
  const int lane = threadIdx.x & 31;
  const int wave = threadIdx.x >> 5;
  const int half = lane >> 4;
  const int lr   = lane & 15;
  const int m0 = (blockIdx.x * 8 + wave) * 16;
  const int row = m0 + lr;

  const int g = row / LSEQ, l = row % LSEQ;
  const int dir = g >> 1, b = g & 1;
  const int r = l / WSZ, s = l % WSZ;
  const float* arow = xmconv + dir_off(dir, b, r, s, 0);

  v8f acc[3] = {};
  for (int k0 = 0; k0 < D_INNER; k0 += 4) {
    v2f a = *(const v2f*)(arow + k0 + 2 * half);
    const int kk = k0 + 2 * half;
#pragma unroll
    for (int t = 0; t < 3; ++t) {
      const int n = t * 16 + lr;
      v2f bv;
      bv.x = (n < DT_RANK + 2 * D_STATE) ? Wx[(size_t)kk * 38 + n] : 0.0f;
      bv.y = (n < DT_RANK + 2 * D_STATE) ? Wx[(size_t)(kk + 1) * 38 + n] : 0.0f;
      acc[t] = wmma4(a, bv, acc[t]);
    }
  }
#pragma unroll
  for (int t = 0; t < 3; ++t) {
    const int pos = xdbl_pos(t * 16 + lr);
#pragma unroll
    for (int v = 0; v < 8; ++v) {
      const int m = m0 + v + 8 * half;
      xdbl[(size_t)m * XDBL_LD + pos] = acc[t][v];
    }
  }
}

// ---------------- 4) dt = softplus(dtr @ W_dt + b_dt), fused with x gather ---
// writes {dt, x} as float2 per (g,l,c) so the scan does one b64 load.
__global__ void k_dtx(const float* __restrict__ xdbl, const float* __restrict__ Wdt,
                      const float* __restrict__ bdt,
                      const float* __restrict__ xmconv,
                      float* __restrict__ dtx) {
  const int idx = blockIdx.x * 256 + threadIdx.x;   // over MG*D_INNER
  const int c = idx % D_INNER;
  const int m = idx / D_INNER;
  const int g = m / LSEQ, l = m % LSEQ;
  const int dir = g >> 1, b = g & 1;
  const int r = l / WSZ, s = l % WSZ;

  const float* xr = xdbl + (size_t)m * XDBL_LD;
  float acc = bdt[c];
#pragma unroll
  for (int rr = 0; rr < DT_RANK; ++rr) acc += xr[rr] * Wdt[rr * D_INNER + c];
  const float dtv = (acc > 20.0f) ? acc : __logf(1.0f + __expf(acc));
  const float xv  = xmconv[dir_off(dir, b, r, s, c)];
  v2f o = { dtv, xv };
  *(v2f*)(dtx + (size_t)idx * 2) = o;
}

// ---------------- 5) selective scan ----------------
// block = 16 channels x 16 states; thread owns h(c,n) in a register.
// Serial loop: one b64 {dt,x}, one b64 {B,C}, exp, fma chain, 16-lane dot.
__global__ void k_scan(const float* __restrict__ dtx,
                       const float* __restrict__ xdbl,
                       const float* __restrict__ Alog,
                       const float* __restrict__ Dsk,
                       float* __restrict__ ys) {
  const int g  = blockIdx.y;
  const int ct = blockIdx.x;
  const int n  = threadIdx.x & 15;
  const int ty = threadIdx.x >> 4;
  const int c  = ct * 16 + ty;

  const float A  = -__expf(Alog[c * D_STATE + n]);
  const float Dv = Dsk[c];
  const float* pbc = xdbl + (size_t)g * LSEQ * XDBL_LD + DT_RANK + 2 * n;
  const float* pdx = dtx + ((size_t)g * LSEQ * D_INNER + c) * 2;
  float*       yp  = ys  + (size_t)g * LSEQ * D_INNER + c;

  float h = 0.0f;
  v2f bc = *(const v2f*)pbc;          // {B_n, C_n} at l=0
  v2f dx = *(const v2f*)pdx;          // {dt, x}   at l=0
#pragma unroll 4
  for (int l = 0; l < LSEQ; ++l) {
    const v2f bcc = bc, dxc = dx;
    if (l + 1 < LSEQ) {               // 1-deep operand prefetch into registers
      bc = *(const v2f*)(pbc + (size_t)(l + 1) * XDBL_LD);
      dx = *(const v2f*)(pdx + (size_t)(l + 1) * (D_INNER * 2));
    }
    if ((l & 3) == 0) {               // stream-ahead prefetch (global_prefetch_b8)
      __builtin_prefetch(pbc + (size_t)(l + 32) * XDBL_LD, 0, 0);
      __builtin_prefetch(pdx + (size_t)(l + 32) * (D_INNER * 2), 0, 0);
    }
    h = __expf(dxc.x * A) * h + dxc.x * bcc.x * dxc.y;
    float y = h * bcc.y;
    y += __shfl_xor(y, 1);
    y += __shfl_xor(y, 2);
    y += __shfl_xor(y, 4);
    y += __shfl_xor(y, 8);
    if (n == 0) yp[(size_t)l * D_INNER] = y + Dv * dxc.y;
  }
}

// ---------------- 6) y_pre = silu(z) * sum_k ys ----------------
__global__ void k_combine(const float* __restrict__ ys, const float* __restrict__ z,
                          float* __restrict__ ypre) {
  const int idx = blockIdx.x * 256 + threadIdx.x;   // over M1*D_INNER
  const int c = idx % D_INNER;
  const int m = idx / D_INNER;            // m = b*LSEQ + l
  const int b = m / LSEQ, l = m % LSEQ;
  float acc = 0.0f;
#pragma unroll
  for (int k = 0; k < 4; ++k)
    acc += ys[((size_t)(k * BSZ + b) * LSEQ + l) * D_INNER + c];
  const float zv = z[idx];
  ypre[idx] = (zv / (1.0f + __expf(-zv))) * acc;
}

// ---------------- 7) out = y_pre @ W_out (6272x192 @ 192x96) ----------------
__global__ void k_gemm_out(const float* __restrict__ ypre,
                           const float* __restrict__ Wout,
                           float* __restrict__ out) {
  const int lane = threadIdx.x & 31;
  const int wave = threadIdx.x >> 5;
  const int half = lane >> 4;
  const int lr   = lane & 15;
  const int m0 = (blockIdx.x * 8 + wave) * 16;

  v8f acc[6] = {};
  const float* arow = ypre + (size_t)(m0 + lr) * D_INNER;
  for (int k0 = 0; k0 < D_INNER; k0 += 4) {
    v2f a = *(const v2f*)(arow + k0 + 2 * half);
    const int kk = k0 + 2 * half;
    const float* bp = Wout + (size_t)kk * D_MODEL + lr;
#pragma unroll
    for (int t = 0; t < 6; ++t) {
      v2f bv = { bp[t * 16], bp[D_MODEL + t * 16] };
      acc[t] = wmma4(a, bv, acc[t]);
    }
  }
#pragma unroll
  for (int t = 0; t < 6; ++t) {
    const int col = t * 16 + lr;
#pragma unroll
    for (int v = 0; v < 8; ++v) {
      const int m = m0 + v + 8 * half;
      out[(size_t)m * D_MODEL + col] = acc[t][v];
    }
  }
}

// ---------------- launch ----------------
extern "C" void kernel_launch(void* const* d_in, const int* in_sizes, int n_in,
                              void* d_out, int out_size, void* d_ws, size_t ws_size,
                              hipStream_t stream) {
  const float* x     = (const float*)d_in[0];
  const float* Win   = (const float*)d_in[1];
  const float* convw = (const float*)d_in[2];
  const float* convb = (const float*)d_in[3];
  const float* Wx    = (const float*)d_in[4];
  const float* Wdt   = (const float*)d_in[5];
  const float* bdt   = (const float*)d_in[6];
  const float* Alog  = (const float*)d_in[7];
  const float* Dsk   = (const float*)d_in[8];
  const float* Wout  = (const float*)d_in[9];
  float* out = (float*)d_out;

  float* ws = (float*)d_ws;
  size_t o = 0;
  float* xm_raw  = ws + o; o += (size_t)M1 * D_INNER;        // 6272*192
  float* zbuf    = ws + o; o += (size_t)M1 * D_INNER;
  float* xm_conv = ws + o; o += (size_t)M1 * D_INNER;
  float* xdbl    = ws + o; o += (size_t)MG * XDBL_LD;        // 25088*48
  float* dtx     = ws + o; o += (size_t)MG * D_INNER * 2;    // {dt,x} pairs
  float* ysb     = ws + o; o += (size_t)MG * D_INNER;
  float* ypre    = ws + o; o += (size_t)M1 * D_INNER;
  (void)ws_size; (void)in_sizes; (void)n_in; (void)out_size;

  const dim3 blk(256);
  k_gemm_in   <<<dim3(6, (M1 / 16) / 8), blk, 0, stream>>>(x, Win, xm_raw, zbuf);
  k_dwconv    <<<dim3((M1 * D_INNER) / 256), blk, 0, stream>>>(xm_raw, convw, convb, xm_conv);
  k_gemm_xproj<<<dim3((MG / 16) / 8), blk, 0, stream>>>(xm_conv, Wx, xdbl);
  k_dtx       <<<dim3((MG * D_INNER) / 256), blk, 0, stream>>>(xdbl, Wdt, bdt, xm_conv, dtx);
  k_scan      <<<dim3(D_INNER / 16, GRP), blk, 0, stream>>>(dtx, xdbl, Alog, Dsk, ysb);
  k_combine   <<<dim3((M1 * D_INNER) / 256), blk, 0, stream>>>(ysb, zbuf, ypre);
  k_gemm_out  <<<dim3((M1 / 16) / 8), blk, 0, stream>>>(ypre, Wout, out);
}